// VQC_71201967833235
// MI455X (gfx1250) — compile-verified
//
#include <hip/hip_runtime.h>

// ============================================================================
// 18-qubit VQC statevector simulator for MI455X (gfx1250, wave32).
//
// Layer algebra:  layer = GrayPerm ∘ (⊗_q U_q)   where the full adjacent-CNOT
// chain collapses to the index map  out[i] = in[i ^ (i>>1)]  (binary->Gray).
// Rotations are fused 4 qubits at a time into 16x16 complex gates applied via
// V_WMMA_F32_16X16X4_F32 (full fp32 precision, 16 WMMAs per 16-fiber tile).
// The remaining 2-qubit gate (qubits 16,17 = lowest two index bits) is folded
// into the permutation pass, which maps aligned quads onto aligned quads.
//
// d_ws layout: state bufA (128MB) | state bufB (128MB) | gates (~49KB) | partials
// ============================================================================

typedef __attribute__((ext_vector_type(2))) float v2f;
typedef __attribute__((ext_vector_type(8))) float v8f;

#define NLAYERS 4
#define NQ      18
#define NBATCH  64
#define NSTATE  (1 << NQ)          // 262144 amplitudes per batch element
#define TOTAL   (NBATCH * NSTATE)  // 2^24 amplitudes

// ---------------------------------------------------------------------------
// Single-qubit unitary from (phi, theta, omega) — matches reference _rot_matrix
//   a = exp(-0.5i(phi+omega)), b = exp(0.5i(phi-omega)), c=cos(t/2), s=sin(t/2)
//   U = [[a c, -b s], [conj(b) s, conj(a) c]]
// ---------------------------------------------------------------------------
__device__ inline void rot_u(float phi, float theta, float omega,
                             float ur[2][2], float ui[2][2]) {
  float c  = cosf(0.5f * theta);
  float s  = sinf(0.5f * theta);
  float ap = -0.5f * (phi + omega);
  float bp =  0.5f * (phi - omega);
  float ar = cosf(ap), ai = sinf(ap);
  float br = cosf(bp), bi = sinf(bp);
  ur[0][0] =  ar * c;  ui[0][0] =  ai * c;
  ur[0][1] = -br * s;  ui[0][1] = -bi * s;
  ur[1][0] =  br * s;  ui[1][0] = -bi * s;   // conj(b)*s
  ur[1][1] =  ar * c;  ui[1][1] = -ai * c;   // conj(a)*c
}

// ---------------------------------------------------------------------------
// Build fused gates.
//  blocks 0..15 : (layer l = bid>>2, block b = bid&3) 16x16 gate = U_{4b}⊗..⊗U_{4b+3}
//                 stored as 3 planes of 256 floats: Re, Im, -Im (row-major [out][in])
//  blocks 16..19: layer l = bid-16, 4x4 gate = U_16 ⊗ U_17, stored as 16 x float2
// ---------------------------------------------------------------------------
__global__ __launch_bounds__(256)
void build_gates(const float* __restrict__ w, float* __restrict__ G16,
                 v2f* __restrict__ G4) {
  int bid = blockIdx.x;
  int tid = threadIdx.x;
  if (bid < 16) {
    int l = bid >> 2, b = bid & 3;
    int mp = tid >> 4, m = tid & 15;       // output row, input col
    float re = 1.0f, im = 0.0f;
    #pragma unroll
    for (int t = 0; t < 4; ++t) {
      int q = 4 * b + t;                   // qubit; bit (3-t) of m
      const float* wq = w + (l * NQ + q) * 3;
      float ur[2][2], ui[2][2];
      rot_u(wq[0], wq[1], wq[2], ur, ui);
      int rb = (mp >> (3 - t)) & 1, cb = (m >> (3 - t)) & 1;
      float xr = ur[rb][cb], xi = ui[rb][cb];
      float nr = re * xr - im * xi;
      float ni = re * xi + im * xr;
      re = nr; im = ni;
    }
    float* base = G16 + bid * 768;
    base[tid]       = re;
    base[256 + tid] = im;
    base[512 + tid] = -im;                 // pre-negated plane (F32 WMMA has no A-neg)
  } else if (tid < 16) {
    int l = bid - 16;
    int mp = tid >> 2, m = tid & 3;        // bit1 = qubit16, bit0 = qubit17
    float re = 1.0f, im = 0.0f;
    #pragma unroll
    for (int t = 0; t < 2; ++t) {
      int q = 16 + t;
      const float* wq = w + (l * NQ + q) * 3;
      float ur[2][2], ui[2][2];
      rot_u(wq[0], wq[1], wq[2], ur, ui);
      int rb = (mp >> (1 - t)) & 1, cb = (m >> (1 - t)) & 1;
      float xr = ur[rb][cb], xi = ui[rb][cb];
      float nr = re * xr - im * xi;
      float ni = re * xi + im * xr;
      re = nr; im = ni;
    }
    v2f g; g.x = re; g.y = im;
    G4[l * 16 + tid] = g;
  }
}

// ---------------------------------------------------------------------------
// Initialize basis state:  idx(b) = sum_{t<9} x[b,t] * 2^(17-t)
// ---------------------------------------------------------------------------
__global__ __launch_bounds__(256)
void init_state(v2f* __restrict__ s, const int* __restrict__ x) {
  __shared__ int sidx;
  int gid = blockIdx.x * 256 + threadIdx.x;
  int batch = gid >> NQ;                   // 256 | 2^18, block is single-batch
  if (threadIdx.x == 0) {
    int idx = 0;
    #pragma unroll
    for (int t = 0; t < 9; ++t) idx += x[batch * 9 + t] << (17 - t);
    sidx = idx;
  }
  __syncthreads();
  int i = gid & (NSTATE - 1);
  v2f v; v.x = (i == sidx) ? 1.0f : 0.0f; v.y = 0.0f;
  s[gid] = v;
}

// ---------------------------------------------------------------------------
// In-place 16x16 complex gate on qubit block with element stride 2^p.
// One wave = one 16(fiber) x 16(m) tile, 4 tiles per wave.
//   D = A x B + C with V_WMMA_F32_16X16X4_F32 (4 K-chunks x 4 real products).
//   A (gate, 16x4 f32, 2 VGPR): lane L -> M = L&15, K pair = 4c + 2*(L>>4)+{0,1}
//   B (state, 4x16 f32, 2 VGPR): lane L -> N = L&15 (fiber), same K-pair split
//   C/D (16x16 f32, 8 VGPR):     lane L -> N = L&15, M = vgpr + 8*(L>>4)
// ---------------------------------------------------------------------------
__global__ __launch_bounds__(256)
void gate16_pass(v2f* __restrict__ state, const float* __restrict__ G, int p) {
  const int lane = threadIdx.x & 31;
  const int n    = lane & 15;              // fiber column / gate output row lane
  const int kh   = lane >> 4;              // K-pair selector
  const int wave = blockIdx.x * 8 + (threadIdx.x >> 5);

  // Gate operand registers, loaded once, reused for 4 tiles.
  const float* Gr = G;
  const float* Gi = G + 256;
  const float* Gn = G + 512;               // = -Gi
  v2f Ar[4], Ai[4], An[4];
  #pragma unroll
  for (int c = 0; c < 4; ++c) {
    int k0 = 4 * c + 2 * kh;
    Ar[c].x = Gr[n * 16 + k0]; Ar[c].y = Gr[n * 16 + k0 + 1];
    Ai[c].x = Gi[n * 16 + k0]; Ai[c].y = Gi[n * 16 + k0 + 1];
    An[c].x = Gn[n * 16 + k0]; An[c].y = Gn[n * 16 + k0 + 1];
  }

  const int mask = (1 << p) - 1;
  #pragma unroll
  for (int it = 0; it < 4; ++it) {
    int t     = wave * 4 + it;             // tile id, 65536 total
    int batch = t >> 10;
    int f     = ((t & 1023) << 4) | n;     // fiber index (14 bits)
    int base  = (batch << NQ) + ((f >> p) << (p + 4)) + (f & mask);

    v8f Dr = {0.f,0.f,0.f,0.f,0.f,0.f,0.f,0.f};
    v8f Di = {0.f,0.f,0.f,0.f,0.f,0.f,0.f,0.f};
    #pragma unroll
    for (int c = 0; c < 4; ++c) {
      int k0 = 4 * c + 2 * kh;
      v2f c0 = state[base + (k0 << p)];
      v2f c1 = state[base + ((k0 + 1) << p)];
      v2f Br; Br.x = c0.x; Br.y = c1.x;
      v2f Bi; Bi.x = c0.y; Bi.y = c1.y;
      // Dr += Ar*Br - Ai*Bi ; Di += Ar*Bi + Ai*Br
      Dr = __builtin_amdgcn_wmma_f32_16x16x4_f32(false, Ar[c], false, Br, (short)0, Dr, false, false);
      Dr = __builtin_amdgcn_wmma_f32_16x16x4_f32(false, An[c], false, Bi, (short)0, Dr, false, false);
      Di = __builtin_amdgcn_wmma_f32_16x16x4_f32(false, Ar[c], false, Bi, (short)0, Di, false, false);
      Di = __builtin_amdgcn_wmma_f32_16x16x4_f32(false, Ai[c], false, Br, (short)0, Di, false, false);
    }
    #pragma unroll
    for (int r = 0; r < 8; ++r) {
      int M = r + 8 * kh;
      v2f o; o.x = Dr[r]; o.y = Di[r];
      state[base + (M << p)] = o;
    }
  }
}

// ---------------------------------------------------------------------------
// Apply 4x4 gate on qubits 16,17 (lowest 2 index bits) then the Gray
// permutation  out[i] = rot[i ^ (i>>1)].  Aligned quads map to aligned quads:
//   out[i0+k] = rot[q ^ gray(k)],  q = i0 ^ (i0>>1),  gray = {0,1,3,2}.
// Ping-pong src -> dst.
// ---------------------------------------------------------------------------
__global__ __launch_bounds__(256)
void perm_g4_pass(const v2f* __restrict__ src, v2f* __restrict__ dst,
                  const v2f* __restrict__ G4) {
  __shared__ v2f g[16];
  if (threadIdx.x < 16) g[threadIdx.x] = G4[threadIdx.x];
  __syncthreads();

  int Q     = blockIdx.x * 256 + threadIdx.x;  // 2^22 quads
  int batch = Q >> 16;
  int i0    = (Q & 65535) << 2;                // 18-bit, quad aligned
  int q     = i0 ^ (i0 >> 1);
  int jb    = q & ~3;
  int boff  = batch << NQ;

  v2f sv[4];
  #pragma unroll
  for (int m = 0; m < 4; ++m) sv[m] = src[boff + jb + m];

  v2f r[4];
  #pragma unroll
  for (int mp = 0; mp < 4; ++mp) {
    float rr = 0.f, ri = 0.f;
    #pragma unroll
    for (int m = 0; m < 4; ++m) {
      v2f gg = g[mp * 4 + m];
      rr += gg.x * sv[m].x - gg.y * sv[m].y;
      ri += gg.x * sv[m].y + gg.y * sv[m].x;
    }
    r[mp].x = rr; r[mp].y = ri;
  }

  int qh = q & 3;
  dst[boff + i0 + 0] = r[qh ^ 0];
  dst[boff + i0 + 1] = r[qh ^ 1];
  dst[boff + i0 + 2] = r[qh ^ 3];   // gray(2)=3
  dst[boff + i0 + 3] = r[qh ^ 2];   // gray(3)=2
}

// ---------------------------------------------------------------------------
// Readout: ev_a(b) = sum_j |s[b,j]|^2 * (bit_{17-a}(j) ? -1 : +1), a = 0..3.
// Deterministic two-stage reduction (no float atomics).
// ---------------------------------------------------------------------------
__global__ __launch_bounds__(256)
void readout_partial(const v2f* __restrict__ s, float* __restrict__ part) {
  int batch = blockIdx.x >> 5, slice = blockIdx.x & 31;
  int base  = (batch << NQ) + (slice << 13);   // 8192 elements per slice
  float a0 = 0.f, a1 = 0.f, a2 = 0.f, a3 = 0.f;
  for (int e = threadIdx.x; e < 8192; e += 256) {
    int j = (slice << 13) + e;
    v2f v = s[base + e];
    float pp = v.x * v.x + v.y * v.y;
    a0 += ((j >> 17) & 1) ? -pp : pp;
    a1 += ((j >> 16) & 1) ? -pp : pp;
    a2 += ((j >> 15) & 1) ? -pp : pp;
    a3 += ((j >> 14) & 1) ? -pp : pp;
  }
  __shared__ float red[256][4];
  red[threadIdx.x][0] = a0; red[threadIdx.x][1] = a1;
  red[threadIdx.x][2] = a2; red[threadIdx.x][3] = a3;
  __syncthreads();
  for (int off = 128; off > 0; off >>= 1) {
    if (threadIdx.x < (unsigned)off)
      #pragma unroll
      for (int a = 0; a < 4; ++a) red[threadIdx.x][a] += red[threadIdx.x + off][a];
    __syncthreads();
  }
  if (threadIdx.x < 4) part[blockIdx.x * 4 + threadIdx.x] = red[0][threadIdx.x];
}

__global__ void readout_final(const float* __restrict__ part, float* __restrict__ out) {
  int batch = blockIdx.x;
  int a = threadIdx.x;
  if (a < 4) {
    float s = 0.f;
    for (int sl = 0; sl < 32; ++sl) s += part[(batch * 32 + sl) * 4 + a];
    out[batch * 4 + a] = s;
  }
}

// ---------------------------------------------------------------------------
extern "C" void kernel_launch(void* const* d_in, const int* in_sizes, int n_in,
                              void* d_out, int out_size, void* d_ws, size_t ws_size,
                              hipStream_t stream) {
  const float* w = (const float*)d_in[0];   // (4,18,3) f32
  const int*   x = (const int*)d_in[1];     // (64,9)  i32
  float*     out = (float*)d_out;           // (64,4)  f32

  v2f*   bufA = (v2f*)d_ws;
  v2f*   bufB = bufA + TOTAL;
  float* G16  = (float*)(bufB + TOTAL);     // 16 gates * 768 floats
  v2f*   G4   = (v2f*)(G16 + 16 * 768);     // 4 layers * 16 complex
  float* part = (float*)(G4 + 4 * 16);      // 64*32*4 partials

  init_state<<<TOTAL / 256, 256, 0, stream>>>(bufA, x);
  build_gates<<<20, 256, 0, stream>>>(w, G16, G4);

  v2f* cur = bufA;
  v2f* nxt = bufB;
  for (int l = 0; l < NLAYERS; ++l) {
    for (int b = 0; b < 4; ++b)
      gate16_pass<<<2048, 256, 0, stream>>>(cur, G16 + (l * 4 + b) * 768, 14 - 4 * b);
    perm_g4_pass<<<16384, 256, 0, stream>>>(cur, nxt, G4 + l * 16);
    v2f* t = cur; cur = nxt; nxt = t;
  }

  readout_partial<<<NBATCH * 32, 256, 0, stream>>>(cur, part);
  readout_final<<<NBATCH, 64, 0, stream>>>(part, out);
}